// RoIAlign_43550968381612
// MI455X (gfx1250) — compile-verified
//
#include <hip/hip_runtime.h>
#include <stdint.h>

// ---------------- problem constants (from reference setup_inputs) ----------
#define B_     4
#define C_     256
#define H_     200
#define W_     272
#define HW_    (H_ * W_)          // 54400 (fits in 16 bits -> stride1 low word)
#define NROI   2048
#define PH_    7
#define PW_    7
#define NBIN   (PH_ * PW_)        // 49
#define NSAMP  (NBIN * 4)         // 196 samples (2x2 per bin)
#define SCALE  0.0625f
#define MAXW   34                 // max window extent (6.5*bin + 2 <= 32, padded)
#define CHUNK  4                  // channels per TDM tile
#define NCHUNK (C_ / CHUNK)       // 64

typedef uint32_t u32x4 __attribute__((ext_vector_type(4)));
typedef int      i32x8 __attribute__((ext_vector_type(8)));
typedef int      i32x4 __attribute__((ext_vector_type(4)));

struct Smem {
    float buf[2][CHUNK * MAXW * MAXW]; // double-buffered feature tiles (36,992 B)
    int   off[NSAMP][4];               // per-sample 4 corner offsets (window-rel)
    float wgt[NSAMP][4];               // per-sample 4 bilinear weights (incl. /4)
};

__device__ __forceinline__ uint32_t rfl(uint32_t v) {
    return (uint32_t)__builtin_amdgcn_readfirstlane((int)v);
}

// Build a Tensor-DMA descriptor (D#) for a 3D tile:
//   X = win_w contiguous floats, Y = win_h rows (stride W), Z = CHUNK channels
//   (stride H*W). Window is fully in-bounds, so tensor dims == tile dims.
__device__ __forceinline__ void tdm_load_tile(uint64_t gaddr, uint32_t lds_byte_addr,
                                              uint32_t win_w, uint32_t win_h)
{
    u32x4 g0;
    g0[0] = rfl(1u);                                        // count=1, user mode, no gather
    g0[1] = rfl(lds_byte_addr);                             // lds_addr [63:32]
    g0[2] = rfl((uint32_t)(gaddr & 0xffffffffull));         // global_addr[31:0]
    g0[3] = rfl(((uint32_t)((gaddr >> 32) & 0x1ffffffull))  // global_addr[56:32]
                | (2u << 30));                              // type = 2 ("image")
    i32x8 g1;
    g1[0] = (int)rfl(2u << 16);                             // data_size=2 (4B); wg_mask=0
    g1[1] = (int)rfl(win_w << 16);                          // tensor_dim0[15:0] @ [63:48]
    g1[2] = (int)rfl(win_h << 16);                          // td0 hi=0 | tensor_dim1[15:0]
    g1[3] = (int)rfl(win_w << 16);                          // td1 hi=0 | tile_dim0
    g1[4] = (int)rfl(((uint32_t)CHUNK << 16) | win_h);      // tile_dim1 | tile_dim2
    g1[5] = (int)rfl((uint32_t)W_);                         // tensor_dim0_stride[31:0]
    g1[6] = (int)rfl(((uint32_t)HW_ & 0xffffu) << 16);      // s0 hi=0 | tensor_dim1_stride[15:0]
    g1[7] = (int)rfl((uint32_t)HW_ >> 16);                  // tensor_dim1_stride[47:16] (=0)
    i32x4 g2;
    g2[0] = (int)rfl((uint32_t)CHUNK);                      // tensor_dim2
    g2[1] = 0; g2[2] = 0; g2[3] = 0;
    i32x4 g3; g3[0] = 0; g3[1] = 0; g3[2] = 0; g3[3] = 0;
    i32x8 g4; g4[0] = 0; g4[1] = 0; g4[2] = 0; g4[3] = 0;   // extra operand on this
    g4[4] = 0; g4[5] = 0; g4[6] = 0; g4[7] = 0;             // toolchain: zero-filled
    __builtin_amdgcn_tensor_load_to_lds(g0, g1, g2, g3, g4, 0);
}

__global__ __launch_bounds__(256)
void roialign_tdm_kernel(const float* __restrict__ feat,
                         const float* __restrict__ rois,
                         float* __restrict__ out)
{
    __shared__ Smem s;
    const int n   = blockIdx.x;
    const int tid = threadIdx.x;

    // ---- ROI parameters (uniform across block) ----
    const float* roi = rois + n * 5;
    const int   bidx  = (int)roi[0];
    const float rx1   = roi[1] * SCALE;
    const float ry1   = roi[2] * SCALE;
    const float roi_w = fmaxf(roi[3] * SCALE - rx1, 1.0f);
    const float roi_h = fmaxf(roi[4] * SCALE - ry1, 1.0f);
    const float bin_w = roi_w * (1.0f / PW_);
    const float bin_h = roi_h * (1.0f / PH_);

    // ---- feature window covering all clipped sample corners (uniform) ----
    const float ylo = fminf(fmaxf(ry1 + 0.25f * bin_h, 0.0f), (float)(H_ - 1));
    const float yhi = fminf(fmaxf(ry1 + 6.75f * bin_h, 0.0f), (float)(H_ - 1));
    const float xlo = fminf(fmaxf(rx1 + 0.25f * bin_w, 0.0f), (float)(W_ - 1));
    const float xhi = fminf(fmaxf(rx1 + 6.75f * bin_w, 0.0f), (float)(W_ - 1));
    const int wy0 = (int)floorf(ylo);
    const int wx0 = (int)floorf(xlo);
    const int wy1 = min((int)floorf(yhi) + 1, H_ - 1);
    const int wx1 = min((int)floorf(xhi) + 1, W_ - 1);
    const int win_h = min(wy1 - wy0 + 1, MAXW);
    const int win_w = min(wx1 - wx0 + 1, MAXW);
    const int chanSize = win_w * win_h;

    // scalar-provable wave-0 predicate (TDM ignores EXEC; branch must be SALU)
    const bool wave0 = (rfl((uint32_t)tid >> 5) == 0u);

    const uint64_t gbase = (uint64_t)(uintptr_t)feat
        + 4ull * ((uint64_t)bidx * (uint64_t)(C_ * HW_)
                  + (uint64_t)wy0 * W_ + (uint64_t)wx0);

    // kick off DMA of channel-chunk 0 while we build sample metadata
    if (wave0) {
        tdm_load_tile(gbase, (uint32_t)(uintptr_t)&s.buf[0][0],
                      (uint32_t)win_w, (uint32_t)win_h);
    }

    // ---- per-sample bilinear metadata (196 samples, one thread each) ----
    if (tid < NSAMP) {
        const int bin = tid >> 2, sub = tid & 3;
        const int ph = bin / PW_, pw = bin % PW_;
        const int iy = sub >> 1,  ix = sub & 1;
        const float yy = ry1 + (float)ph * bin_h + ((float)iy + 0.5f) * bin_h * 0.5f;
        const float xx = rx1 + (float)pw * bin_w + ((float)ix + 0.5f) * bin_w * 0.5f;
        const bool valid = (yy >= -1.0f) && (yy <= (float)H_) &&
                           (xx >= -1.0f) && (xx <= (float)W_);
        const float yc = fminf(fmaxf(yy, 0.0f), (float)(H_ - 1));
        const float xc = fminf(fmaxf(xx, 0.0f), (float)(W_ - 1));
        const int y0  = min((int)floorf(yc), H_ - 1);
        const int x0  = min((int)floorf(xc), W_ - 1);
        const int y1i = min(y0 + 1, H_ - 1);
        const int x1i = min(x0 + 1, W_ - 1);
        const float ly = yc - (float)y0, lx = xc - (float)x0;
        const float hy = 1.0f - ly,      hx = 1.0f - lx;
        const float v  = valid ? 0.25f : 0.0f;   // fold 1/(S*S) average + valid mask
        const int a0 = min(max(y0  - wy0, 0), win_h - 1);
        const int a1 = min(max(y1i - wy0, 0), win_h - 1);
        const int b0 = min(max(x0  - wx0, 0), win_w - 1);
        const int b1 = min(max(x1i - wx0, 0), win_w - 1);
        s.off[tid][0] = a0 * win_w + b0;
        s.off[tid][1] = a0 * win_w + b1;
        s.off[tid][2] = a1 * win_w + b0;
        s.off[tid][3] = a1 * win_w + b1;
        s.wgt[tid][0] = hy * hx * v;
        s.wgt[tid][1] = hy * lx * v;
        s.wgt[tid][2] = ly * hx * v;
        s.wgt[tid][3] = ly * lx * v;
    }
    __syncthreads();

    // ---- hoist this thread's 16 offsets/weights into registers ----
    const int bin = tid % NBIN;        // output bin
    const int cl  = tid / NBIN;        // channel-in-chunk (0..3 for active threads)
    int   roff[16];
    float rwgt[16];
    if (tid < CHUNK * NBIN) {          // 196 active threads
        #pragma unroll
        for (int sub = 0; sub < 4; ++sub)
            #pragma unroll
            for (int k = 0; k < 4; ++k) {
                roff[sub * 4 + k] = s.off[bin * 4 + sub][k] + cl * chanSize;
                rwgt[sub * 4 + k] = s.wgt[bin * 4 + sub][k];
            }
    }

    // ---- pipelined main loop over 64 channel chunks ----
    float* outn = out + (size_t)n * (C_ * NBIN);
    for (int ch = 0; ch < NCHUNK; ++ch) {
        if (wave0) {
            __builtin_amdgcn_s_wait_tensorcnt(0);      // chunk ch landed in LDS
            if (ch + 1 < NCHUNK) {                     // prefetch chunk ch+1
                tdm_load_tile(gbase + 4ull * (uint64_t)(ch + 1) * (CHUNK * HW_),
                              (uint32_t)(uintptr_t)&s.buf[(ch + 1) & 1][0],
                              (uint32_t)win_w, (uint32_t)win_h);
            }
        }
        __syncthreads();                               // chunk ch visible to all waves
        if (tid < CHUNK * NBIN) {
            const float* lb = &s.buf[ch & 1][0];
            float acc = 0.0f;
            #pragma unroll
            for (int j = 0; j < 16; ++j)
                acc = fmaf(rwgt[j], lb[roff[j]], acc);
            outn[(ch * CHUNK + cl) * NBIN + bin] = acc;
        }
        __syncthreads();                               // done reading before overwrite
    }
}

extern "C" void kernel_launch(void* const* d_in, const int* in_sizes, int n_in,
                              void* d_out, int out_size, void* d_ws, size_t ws_size,
                              hipStream_t stream)
{
    (void)in_sizes; (void)n_in; (void)out_size; (void)d_ws; (void)ws_size;
    const float* feat = (const float*)d_in[0];   // (4,256,200,272) f32
    const float* rois = (const float*)d_in[1];   // (2048,5) f32
    float*       out  = (float*)d_out;           // (2048,256,7,7) f32
    hipLaunchKernelGGL(roialign_tdm_kernel, dim3(NROI), dim3(256), 0, stream,
                       feat, rois, out);
}